// QuantumAttentionGate_816043786293
// MI455X (gfx1250) — compile-verified
//
#include <hip/hip_runtime.h>

typedef __attribute__((ext_vector_type(2))) float v2f;
typedef __attribute__((ext_vector_type(4))) float f4;
typedef __attribute__((ext_vector_type(8))) float v8f;

#define NQ 8
#define NL 2
#define NSTATE 256       // 2^NQ
#define B_  32
#define NH_ 16
#define T_  1024
#define HD_ 64
#define H_  (NH_*HD_)    // 1024

// ---------------------------------------------------------------------------
// Kernel A: summary[bnh*64 + hd] = mean over T of q[b,nh,t,hd]   (bnh = b*16+nh)
// One block per (b,nh); 256 threads; f4 (B128) coalesced loads.
// ---------------------------------------------------------------------------
__global__ void k_summary(const float* __restrict__ q, float* __restrict__ summary) {
  const int bnh  = blockIdx.x;           // 0..511
  const int tid  = threadIdx.x;          // 0..255
  const int hd4  = (tid & 15) * 4;       // 0,4,...,60
  const int trow = tid >> 4;             // 0..15
  const float* base = q + (size_t)bnh * (size_t)(T_ * HD_);
  float a0 = 0.f, a1 = 0.f, a2 = 0.f, a3 = 0.f;
  for (int t = trow; t < T_; t += 16) {
    const f4 x = *(const f4*)(base + (size_t)t * HD_ + hd4);
    a0 += x.x; a1 += x.y; a2 += x.z; a3 += x.w;
  }
  __shared__ float red[16 * 64];
  red[trow * 64 + hd4 + 0] = a0;
  red[trow * 64 + hd4 + 1] = a1;
  red[trow * 64 + hd4 + 2] = a2;
  red[trow * 64 + hd4 + 3] = a3;
  __syncthreads();
  if (tid < 64) {
    float s = 0.f;
    #pragma unroll
    for (int r = 0; r < 16; ++r) s += red[r * 64 + tid];
    summary[(size_t)bnh * 64 + tid] = s * (1.0f / (float)T_);
  }
}

// ---------------------------------------------------------------------------
// Kernel B1: angles[32,8] = summary[32,1024] @ W_angles^T[1024,8] + b_angles
// Single wave, V_WMMA_F32_16X16X4_F32.  N padded 8->16 with zero B columns.
// A frag (16x4 f32): lane l<16 holds A[M=l][K=k..k+1] in v0..v1, lanes 16..31
// hold K=k+2..k+3.  B frag mirrors with N on lane.  D: VGPR r -> M = r + 8*half.
// ---------------------------------------------------------------------------
__global__ void k_angles(const float* __restrict__ summary,
                         const float* __restrict__ Wa,    // [NQ, H]
                         const float* __restrict__ ba,    // [NQ]
                         float* __restrict__ angles) {    // [32, NQ]
  const int lane = threadIdx.x;     // 0..31
  const int half = lane >> 4;       // 0 -> K0/K1, 1 -> K2/K3
  const int lm   = lane & 15;       // M (for A) / N (for B) within half
  const float bmask = (lm < NQ) ? 1.0f : 0.0f;
  const float* brow = Wa + (size_t)(lm < NQ ? lm : 0) * H_;
  #pragma unroll
  for (int mt = 0; mt < 2; ++mt) {
    v8f acc = {};
    const float* arow = summary + (size_t)(mt * 16 + lm) * H_;
    for (int k = 0; k < H_; k += 4) {
      const int ka = k + half * 2;
      v2f a;  a.x  = arow[ka];         a.y  = arow[ka + 1];
      v2f bb; bb.x = brow[ka] * bmask; bb.y = brow[ka + 1] * bmask;
      acc = __builtin_amdgcn_wmma_f32_16x16x4_f32(
          /*neg_a=*/false, a, /*neg_b=*/false, bb,
          /*c_mod=*/(short)0, acc, /*reuse_a=*/false, /*reuse_b=*/false);
    }
    if (lm < NQ) {
      #pragma unroll
      for (int r = 0; r < 8; ++r) {
        const int dm = mt * 16 + half * 8 + r;      // M index
        angles[dm * NQ + lm] = acc[r] + ba[lm];
      }
    }
  }
}

// ---------------------------------------------------------------------------
// Kernel B2: 8-qubit statevector sim per batch element; scales[b*16+h] output.
// One block (256 threads = one thread per amplitude) per batch; state in LDS.
// Wire w <-> bit (NQ-1-w) of the flat index (axis 0 is MSB in the reference).
// ---------------------------------------------------------------------------
__global__ void k_circuit(const float* __restrict__ angles,   // [32, NQ]
                          const float* __restrict__ qw,       // [NL, NQ, 3]
                          const float* __restrict__ W_post,   // [NH, NQ]
                          const float* __restrict__ b_post,   // [NH]
                          float* __restrict__ scales) {       // [32, NH]
  const int b   = blockIdx.x;
  const int idx = threadIdx.x;   // 0..255
  __shared__ float sre[NSTATE];
  __shared__ float sim_[NSTATE];
  __shared__ float zacc[NQ];

  sre[idx] = (idx == 0) ? 1.0f : 0.0f;
  sim_[idx] = 0.0f;
  if (idx < NQ) zacc[idx] = 0.0f;

  auto ry = [&](float t, int w) {
    const int  mask = 1 << (NQ - 1 - w);
    const float c = cosf(t * 0.5f), s = sinf(t * 0.5f);
    __syncthreads();
    const float ar = sre[idx],        ai = sim_[idx];
    const float pr = sre[idx ^ mask], pi = sim_[idx ^ mask];
    __syncthreads();
    const float sg = (idx & mask) ? s : -s;   // bit0: c*a - s*p ; bit1: c*a + s*p
    sre[idx] = c * ar + sg * pr;
    sim_[idx] = c * ai + sg * pi;
  };
  auto rz = [&](float t, int w) {
    const int  mask = 1 << (NQ - 1 - w);
    const float ph = (idx & mask) ? 0.5f * t : -0.5f * t;  // e^{+-i t/2}
    const float c = cosf(ph), s = sinf(ph);
    __syncthreads();
    const float ar = sre[idx], ai = sim_[idx];
    sre[idx] = ar * c - ai * s;
    sim_[idx] = ar * s + ai * c;
  };
  auto cnot = [&](int c, int t) {
    const int mc = 1 << (NQ - 1 - c), mt = 1 << (NQ - 1 - t);
    __syncthreads();
    const float pr = sre[idx ^ mt], pi = sim_[idx ^ mt];
    __syncthreads();
    if (idx & mc) { sre[idx] = pr; sim_[idx] = pi; }
  };

  // AngleEmbedding (RY)
  for (int i = 0; i < NQ; ++i) ry(angles[b * NQ + i], i);
  // StronglyEntanglingLayers: Rot = RZ(omega) RY(theta) RZ(phi) (phi applied first)
  for (int l = 0; l < NL; ++l) {
    for (int i = 0; i < NQ; ++i) {
      const float phi   = qw[(l * NQ + i) * 3 + 0];
      const float theta = qw[(l * NQ + i) * 3 + 1];
      const float omega = qw[(l * NQ + i) * 3 + 2];
      rz(phi, i); ry(theta, i); rz(omega, i);
    }
    const int r = (l % (NQ - 1)) + 1;
    for (int i = 0; i < NQ; ++i) cnot(i, (i + r) % NQ);
  }

  __syncthreads();
  const float pr = sre[idx] * sre[idx] + sim_[idx] * sim_[idx];
  #pragma unroll
  for (int i = 0; i < NQ; ++i) {
    const float sg = ((idx >> (NQ - 1 - i)) & 1) ? -pr : pr;
    atomicAdd(&zacc[i], sg);                          // ds_add_f32
  }
  __syncthreads();

  if (idx < NH_) {
    float hl = b_post[idx];
    #pragma unroll
    for (int n = 0; n < NQ; ++n) hl += zacc[n] * W_post[idx * NQ + n];
    scales[b * NH_ + idx] = 1.0f + 0.5f * tanhf(hl);
  }
}

// ---------------------------------------------------------------------------
// Kernel C: out = [q*s | k*s | v*s], streaming, B128, NT hints on k/v and all
// stores (write-once); q load stays RT (may hit L2 resident from kernel A).
// ---------------------------------------------------------------------------
__global__ void k_scale(const f4* __restrict__ q, const f4* __restrict__ k,
                        const f4* __restrict__ v, const float* __restrict__ scales,
                        f4* __restrict__ out, long long n4) {
  const long long i = (long long)blockIdx.x * blockDim.x + threadIdx.x;
  if (i >= n4) return;
  const float s = scales[(int)(i >> 14)];   // 65536 floats = 16384 f4 per (b,h)
  f4 a = q[i];
  f4 bb = __builtin_nontemporal_load(k + i);
  f4 cc = __builtin_nontemporal_load(v + i);
  a *= s; bb *= s; cc *= s;
  __builtin_nontemporal_store(a,  out + i);
  __builtin_nontemporal_store(bb, out + n4 + i);
  __builtin_nontemporal_store(cc, out + 2 * n4 + i);
}

// ---------------------------------------------------------------------------
extern "C" void kernel_launch(void* const* d_in, const int* in_sizes, int n_in,
                              void* d_out, int out_size, void* d_ws, size_t ws_size,
                              hipStream_t stream) {
  const float* q        = (const float*)d_in[0];  // [32,16,1024,64]
  const float* kk       = (const float*)d_in[1];
  const float* vv       = (const float*)d_in[2];
  const float* W_angles = (const float*)d_in[3];  // [8,1024]
  const float* b_angles = (const float*)d_in[4];  // [8]
  const float* q_weights= (const float*)d_in[5];  // [2,8,3]
  const float* W_post   = (const float*)d_in[6];  // [16,8]
  const float* b_post   = (const float*)d_in[7];  // [16]
  float* out = (float*)d_out;

  float* ws      = (float*)d_ws;
  float* summary = ws;                       // 32*1024
  float* angles  = ws + (B_ * H_);           // 32*8
  float* scales  = angles + (B_ * NQ);       // 32*16

  // Phase A: reduce q over T  (128 MB read)
  k_summary<<<B_ * NH_, 256, 0, stream>>>(q, summary);

  // Phase B1: angles GEMM via V_WMMA_F32_16X16X4_F32 (single wave)
  k_angles<<<1, 32, 0, stream>>>(summary, W_angles, b_angles, angles);

  // Phase B2: quantum circuit + post head (32 blocks, LDS statevector)
  k_circuit<<<B_, NSTATE, 0, stream>>>(angles, q_weights, W_post, b_post, scales);

  // Phase C: streaming scale of q/k/v (768 MB traffic, bandwidth bound)
  const long long n4 = (long long)B_ * NH_ * T_ * HD_ / 4;  // 8,388,608
  const int blocks = (int)((n4 + 255) / 256);
  k_scale<<<blocks, 256, 0, stream>>>((const f4*)q, (const f4*)kk, (const f4*)vv,
                                      scales, (f4*)out, n4);
}